// GNN_A3TGCN_48653389529155
// MI455X (gfx1250) — compile-verified
//
#include <hip/hip_runtime.h>
#include <math.h>

// ---------------------------------------------------------------------------
// A3TGCN on MI455X (gfx1250). Key algebraic simplifications (exact):
//  * H0 == 0 every period  -> R gate unused; GRU collapses to
//       Hn = (1 - sigmoid(S_t @ M_z + c_z)) * tanh(S_t @ M_h + c_h)
//    with M_g = W_g @ L_g[0:32,:]  (8x32),  c_g = b_g @ L_g[0:32,:] + lb_g
//  * GCN aggregation is linear & period/gate independent:
//       S = (A_norm) @ X  done ONCE over all 96 raw features, self-loop
//       term dinv^2 * x added at consume time.
// Dense math uses V_WMMA_F32_16X16X4_F32 (full f32 precision).
// ---------------------------------------------------------------------------

typedef __attribute__((ext_vector_type(2))) float v2f;
typedef __attribute__((ext_vector_type(8))) float v8f;

#define NNODES 20000
#define NEDGES 640000
#define NF     8
#define NP     12
#define NH     32
#define FPN    96            // NF * NP features per node

// workspace layout (floats)
#define OFF_DEG   0
#define OFF_DINV  (NNODES)
#define OFF_S     (2 * NNODES)
#define OFF_MZ    (OFF_S + NNODES * FPN)
#define OFF_MH    (OFF_MZ + NF * NH)
#define OFF_CZ    (OFF_MH + NF * NH)
#define OFF_CH    (OFF_CZ + NH)
#define OFF_PR    (OFF_CH + NH)
#define WS_FLOATS (OFF_PR + NP)   // ~7.84 MB

// --------------------------- small helper kernels ---------------------------

__global__ void k_zero(float* ws, int n) {
  int i = blockIdx.x * blockDim.x + threadIdx.x;
  if (i < n) ws[i] = 0.0f;
}

__global__ void k_degree(const int* __restrict__ ei, float* __restrict__ deg) {
  int e = blockIdx.x * blockDim.x + threadIdx.x;
  if (e < NEDGES) atomicAdd(deg + ei[NEDGES + e], 1.0f);
}

__global__ void k_dinv(float* __restrict__ ws) {
  int n = blockIdx.x * blockDim.x + threadIdx.x;
  if (n < NNODES) ws[OFF_DINV + n] = rsqrtf(ws[OFF_DEG + n] + 1.0f);
}

// Fold gate weights + softmax(attention). One block of 640 threads.
__global__ void k_prep(const float* __restrict__ Wz, const float* __restrict__ bz,
                       const float* __restrict__ Wh, const float* __restrict__ bh,
                       const float* __restrict__ Lz, const float* __restrict__ lbz,
                       const float* __restrict__ Lh, const float* __restrict__ lbh,
                       const float* __restrict__ att, float* __restrict__ ws) {
  int tid = threadIdx.x;
  if (tid < 2 * NF * NH) {                 // M_g[f][o] = sum_k W_g[f][k] * L_g[k][o]
    int g = tid >> 8, rem = tid & 255;
    int f = rem >> 5, o = rem & 31;
    const float* W = g ? Wh : Wz;
    const float* L = g ? Lh : Lz;
    float s = 0.0f;
    for (int k = 0; k < NH; ++k) s += W[f * NH + k] * L[k * NH + o];
    ws[(g ? OFF_MH : OFF_MZ) + f * NH + o] = s;
  } else if (tid < 2 * NF * NH + 2 * NH) { // c_g[o] = b_g @ L_g[:32] + lb_g
    int r = tid - 2 * NF * NH;
    int g = r >> 5, o = r & 31;
    const float* b  = g ? bh  : bz;
    const float* L  = g ? Lh  : Lz;
    const float* lb = g ? lbh : lbz;
    float s = lb[o];
    for (int k = 0; k < NH; ++k) s += b[k] * L[k * NH + o];
    ws[(g ? OFF_CH : OFF_CZ) + o] = s;
  } else if (tid == 2 * NF * NH + 2 * NH) { // softmax over 12 periods
    float mx = att[0];
    for (int p = 1; p < NP; ++p) mx = fmaxf(mx, att[p]);
    float sum = 0.0f, e[NP];
    for (int p = 0; p < NP; ++p) { e[p] = __expf(att[p] - mx); sum += e[p]; }
    float inv = 1.0f / sum;
    for (int p = 0; p < NP; ++p) ws[OFF_PR + p] = e[p] * inv;
  }
}

// One edge pass over all 96 raw features: S[dst] += dinv[src]*dinv[dst] * x[src]
// Thread = (edge, 4-feature chunk): float4 load + 4 global f32 atomics.
__global__ __launch_bounds__(256) void k_scatter(const float* __restrict__ x,
                                                 const int* __restrict__ ei,
                                                 float* __restrict__ ws) {
  long long gid = (long long)blockIdx.x * blockDim.x + threadIdx.x;
  if (gid >= (long long)NEDGES * (FPN / 4)) return;
  int e = (int)(gid / (FPN / 4));
  int c = (int)(gid % (FPN / 4));
  int src = ei[e];
  int dst = ei[NEDGES + e];
  float norm = ws[OFF_DINV + src] * ws[OFF_DINV + dst];
  const float4 v = *(const float4*)(x + (long long)src * FPN + c * 4);
  float* Sd = ws + OFF_S + (long long)dst * FPN + c * 4;
  atomicAdd(Sd + 0, norm * v.x);
  atomicAdd(Sd + 1, norm * v.y);
  atomicAdd(Sd + 2, norm * v.z);
  atomicAdd(Sd + 3, norm * v.w);
}

// --------------------------- WMMA node kernel -------------------------------
// One wave32 per 16-node tile. 10 waves / block, 125 blocks = 1250 tiles.
// Per period t, gate g, half h (cols h*16..h*16+15):
//   D(16x16) = chain of two V_WMMA_F32_16X16X4_F32 over K=8.
__global__ __launch_bounds__(320) void k_node(const float* __restrict__ x,
                                              const float* __restrict__ ws,
                                              const float* __restrict__ linW,
                                              const float* __restrict__ linb,
                                              float* __restrict__ out) {
  __shared__ float lds[10][16 * FPN];      // 60 KB of the 320 KB WGP LDS
  const int wid  = threadIdx.x >> 5;
  const int lane = threadIdx.x & 31;
  const int tile = blockIdx.x * 10 + wid;  // always < 1250
  const int base = tile * 16;
  float* my = lds[wid];

  const float* S    = ws + OFF_S;
  const float* dinv = ws + OFF_DINV;
  const float* Mz   = ws + OFF_MZ;
  const float* Mh   = ws + OFF_MH;
  const float* Cz   = ws + OFF_CZ;
  const float* Ch   = ws + OFF_CH;
  const float* Pr   = ws + OFF_PR;

  // Stage S tile (+ self-loop dinv^2 * x) into LDS: [node 0..15][fp 0..95]
  for (int i = lane; i < 16 * FPN; i += 32) {
    int n = i / FPN, fp = i - n * FPN;
    int node = base + n;
    float dv = dinv[node];
    my[i] = S[(long long)node * FPN + fp] + dv * dv * x[(long long)node * FPN + fp];
  }
  __syncthreads();

  const int m    = lane & 15;        // A row / D column index
  const int col  = lane & 15;
  const int half = lane >> 4;        // lane group 0: K{0,1}; group 1: K{2,3}
  const int koff = half * 2;

  // B operands (M_z / M_h halves) resident in registers.
  v2f Bz[2][2], Bh[2][2];
  for (int h = 0; h < 2; ++h)
    for (int c = 0; c < 2; ++c) {
      int kb = c * 4 + koff;
      int cc = h * 16 + col;
      v2f bz_, bh_;
      bz_.x = Mz[kb * NH + cc];  bz_.y = Mz[(kb + 1) * NH + cc];
      bh_.x = Mh[kb * NH + cc];  bh_.y = Mh[(kb + 1) * NH + cc];
      Bz[h][c] = bz_;  Bh[h][c] = bh_;
    }
  const float czv[2] = { Cz[col], Cz[16 + col] };
  const float chv[2] = { Ch[col], Ch[16 + col] };

  v8f Hacc0 = {}, Hacc1 = {};

  for (int t = 0; t < NP; ++t) {
    const float pt = Pr[t];
    v2f A[2];
    for (int c = 0; c < 2; ++c) {
      int kb = c * 4 + koff;
      v2f a;
      a.x = my[m * FPN + kb * NP + t];
      a.y = my[m * FPN + (kb + 1) * NP + t];
      A[c] = a;
    }
    for (int h = 0; h < 2; ++h) {
      v8f z = {};
      z = __builtin_amdgcn_wmma_f32_16x16x4_f32(false, A[0], false, Bz[h][0],
                                                (short)0, z, false, false);
      z = __builtin_amdgcn_wmma_f32_16x16x4_f32(false, A[1], false, Bz[h][1],
                                                (short)0, z, false, false);
      v8f ht = {};
      ht = __builtin_amdgcn_wmma_f32_16x16x4_f32(false, A[0], false, Bh[h][0],
                                                 (short)0, ht, false, false);
      ht = __builtin_amdgcn_wmma_f32_16x16x4_f32(false, A[1], false, Bh[h][1],
                                                 (short)0, ht, false, false);
      v8f acc = h ? Hacc1 : Hacc0;
      for (int v = 0; v < 8; ++v) {
        float sig = 1.0f / (1.0f + __expf(-(z[v] + czv[h])));
        float e2  = __expf(2.0f * (ht[v] + chv[h]));
        float th  = (e2 - 1.0f) / (e2 + 1.0f);
        acc[v] += pt * (1.0f - sig) * th;
      }
      if (h) Hacc1 = acc; else Hacc0 = acc;
    }
  }

  // Re-stage relu(H) (16x32) into LDS so it can be read in A-operand layout.
  __syncthreads();
  for (int v = 0; v < 8; ++v) {
    int row = v + half * 8;            // C/D layout: row = vgpr + 8*(lane>=16)
    my[row * NH + col]      = fmaxf(Hacc0[v], 0.0f);
    my[row * NH + 16 + col] = fmaxf(Hacc1[v], 0.0f);
  }
  __syncthreads();

  // Readout: relu(H)(16x32) @ lin_W(32x12, zero-padded to 16 cols), K in 8 chunks.
  v8f oacc = {};
  for (int c = 0; c < 8; ++c) {
    int kb = c * 4 + koff;
    v2f a, b;
    a.x = my[m * NH + kb];
    a.y = my[m * NH + kb + 1];
    b.x = (col < NP) ? linW[kb * NP + col]       : 0.0f;
    b.y = (col < NP) ? linW[(kb + 1) * NP + col] : 0.0f;
    oacc = __builtin_amdgcn_wmma_f32_16x16x4_f32(false, a, false, b,
                                                 (short)0, oacc, false, false);
  }
  if (col < NP) {
    float bias = linb[col];
    for (int v = 0; v < 8; ++v) {
      int node = base + v + half * 8;
      out[node * NP + col] = oacc[v] + bias;
    }
  }
}

// ------------------------------- launcher -----------------------------------

extern "C" void kernel_launch(void* const* d_in, const int* in_sizes, int n_in,
                              void* d_out, int out_size, void* d_ws, size_t ws_size,
                              hipStream_t stream) {
  const float* x    = (const float*)d_in[0];
  const int*   ei   = (const int*)  d_in[1];
  const float* Wz   = (const float*)d_in[2];
  const float* bz   = (const float*)d_in[3];
  // d_in[4], d_in[5] = W_r, b_r  (R gate mathematically unused: H0 == 0)
  const float* Wh   = (const float*)d_in[6];
  const float* bh   = (const float*)d_in[7];
  const float* Lz   = (const float*)d_in[8];
  const float* lbz  = (const float*)d_in[9];
  // d_in[10], d_in[11] = L_r, lb_r (unused)
  const float* Lh   = (const float*)d_in[12];
  const float* lbh  = (const float*)d_in[13];
  const float* att  = (const float*)d_in[14];
  const float* linW = (const float*)d_in[15];
  const float* linb = (const float*)d_in[16];
  float* ws  = (float*)d_ws;
  float* out = (float*)d_out;

  // 1) zero deg + dinv + S regions (ws is poisoned by the harness)
  int zcount = OFF_S + NNODES * FPN;
  k_zero<<<(zcount + 255) / 256, 256, 0, stream>>>(ws, zcount);
  // 2) degrees, 3) dinv
  k_degree<<<(NEDGES + 255) / 256, 256, 0, stream>>>(ei, ws + OFF_DEG);
  k_dinv<<<(NNODES + 255) / 256, 256, 0, stream>>>(ws);
  // 4) fold weights + softmax
  k_prep<<<1, 640, 0, stream>>>(Wz, bz, Wh, bh, Lz, lbz, Lh, lbh, att, ws);
  // 5) single edge pass over all 96 features
  long long sthreads = (long long)NEDGES * (FPN / 4);
  k_scatter<<<(int)((sthreads + 255) / 256), 256, 0, stream>>>(x, ei, ws);
  // 6) WMMA node compute + readout (1250 tiles, 10 waves/block)
  k_node<<<125, 320, 0, stream>>>(x, ws, linW, linb, out);
}